// RelativePositionEncoding_78022375899177
// MI455X (gfx1250) — compile-verified
//
#include <hip/hip_runtime.h>
#include <stdint.h>

// RelativePositionEncoding (AF3-style), B=1, N=1024, C_Z=128, FEAT=139.
// Row map inside W: W_res rows [0,66), W_tok rows [66,132), w_ent row 132,
// W_chain rows [133,139).
#define R_MAX   32
#define CZ      128
#define FEAT    139
#define SMEM_BYTES (FEAT * CZ * 4)   // 71168 B, fits in 320 KB/WGP LDS

typedef float    f4  __attribute__((ext_vector_type(4)));
typedef unsigned v4u __attribute__((ext_vector_type(4)));
typedef int      v8i __attribute__((ext_vector_type(8)));
typedef int      v4i __attribute__((ext_vector_type(4)));

__global__ __launch_bounds__(256)
void relpos_kernel(const int* __restrict__ asym,
                   const int* __restrict__ resi,
                   const int* __restrict__ ent,
                   const int* __restrict__ sym,
                   const int* __restrict__ tok,
                   const float* __restrict__ W,
                   float* __restrict__ out,
                   int N)
{
    extern __shared__ float smem[];   // FEAT*CZ floats at LDS offset 0

    // ---- Stage W (139x128 fp32) into LDS with one TDM descriptor ----
    // D# per CDNA5 ISA ch.8: 2D tensor, tile == tensor, data_size = 4B.
    if (threadIdx.x == 0) {
        uint64_t ga = (uint64_t)(uintptr_t)W;
        v4u g0;
        g0[0] = 1u;                                          // count=1, user desc
        g0[1] = 0u;                                          // lds_addr = 0
        g0[2] = (unsigned)ga;                                // global_addr[31:0]
        g0[3] = ((unsigned)(ga >> 32) & 0x01FFFFFFu)         // global_addr[56:32]
              | (2u << 30);                                  // type = 2 ("image")
        v8i g1;
        g1[0] = (int)(2u << 16);                             // data_size=2 -> 4B
        g1[1] = (int)((unsigned)CZ   << 16);                 // tensor_dim0 lo16
        g1[2] = (int)((unsigned)FEAT << 16);                 // dim0 hi16=0 | tensor_dim1 lo16
        g1[3] = (int)((unsigned)CZ   << 16);                 // dim1 hi16=0 | tile_dim0=128
        g1[4] = (int)FEAT;                                   // tile_dim1=139 | tile_dim2=0
        g1[5] = (int)CZ;                                     // tensor_dim0_stride lo32 = 128
        g1[6] = (int)((unsigned)(FEAT * CZ) << 16);          // stride0 hi16=0 | stride1 lo16
        g1[7] = 0;                                           // stride1 hi32 = 0
        v4i g2;
        g2[0] = 1;                                           // tensor_dim2 = 1
        g2[1] = 1;                                           // tensor_dim3 = 1 (iterate off)
        g2[2] = FEAT * CZ;                                   // tensor_dim2_stride lo32
        g2[3] = 0;                                           // stride2 hi16 | tile_dim3=0
        v4i g3; g3[0] = 0; g3[1] = 0; g3[2] = 0; g3[3] = 0;  // dims 3/4 unused
        v8i g4 = {0, 0, 0, 0, 0, 0, 0, 0};                   // optional group: unused
        __builtin_amdgcn_tensor_load_to_lds(g0, g1, g2, g3, g4, /*cpol=*/0);
        __builtin_amdgcn_s_wait_tensorcnt(0);                // TENSORcnt == 0
    }
    __syncthreads();

    // ---- Per-pair gather + add, one wave per (i,j) pair ----
    const int i    = blockIdx.x;
    const int lane = threadIdx.x & 31;     // wave32
    const int wv   = threadIdx.x >> 5;     // 8 waves/block
    const int c    = lane << 2;            // 4 channels per lane

    const int ai = asym[i], ri = resi[i], ei = ent[i], si = sym[i], ti = tok[i];

    float* orow = out + (size_t)i * (size_t)N * CZ;

    // w_ent row is the same for every pair: hoist it.
    const f4 r2 = *(const f4*)(smem + 132 * CZ + c);

    for (int j = wv; j < N; j += 8) {
        const int aj = asym[j], rj = resi[j], ej = ent[j], sj = sym[j], tj = tok[j];
        const bool sc = (ai == aj);
        const bool se = (ei == ej);

        int dres, dtok, dch;
        if (sc) {
            int t = ri - rj + R_MAX;
            dres = t < 0 ? 0 : (t > 2 * R_MAX ? 2 * R_MAX : t);
            if (ri == rj) {
                int u = ti - tj + R_MAX;
                dtok = u < 0 ? 0 : (u > 2 * R_MAX ? 2 * R_MAX : u);
            } else {
                dtok = 2 * R_MAX + 1;
            }
        } else {
            dres = 2 * R_MAX + 1;
            dtok = 2 * R_MAX + 1;
        }
        if (se) {
            int v = si - sj + 2;
            dch = v < 0 ? 0 : (v > 4 ? 4 : v);
        } else {
            dch = 5;
        }

        // Conflict-free ds_load_b128: whole wave reads one contiguous 512B row.
        const f4 r0 = *(const f4*)(smem + dres * CZ + c);
        const f4 r1 = *(const f4*)(smem + (66 + dtok) * CZ + c);
        const f4 r3 = *(const f4*)(smem + (133 + dch) * CZ + c);

        const float sef = se ? 1.0f : 0.0f;
        const f4 sv = {sef, sef, sef, sef};
        f4 acc = r0 + r1 + r3 + sv * r2;

        // 512 B contiguous per wave, streamed; never re-read -> NT store.
        __builtin_nontemporal_store(acc, (f4*)(orow + (size_t)j * CZ + c));
    }
}

extern "C" void kernel_launch(void* const* d_in, const int* in_sizes, int n_in,
                              void* d_out, int out_size, void* d_ws, size_t ws_size,
                              hipStream_t stream) {
    const int*   asym = (const int*)d_in[0];
    const int*   resi = (const int*)d_in[1];
    const int*   ent  = (const int*)d_in[2];
    const int*   sym  = (const int*)d_in[3];
    const int*   tok  = (const int*)d_in[4];
    const float* W    = (const float*)d_in[5];
    float*       out  = (float*)d_out;

    const int N = in_sizes[0];   // B*N with B=1 -> 1024

    // 71168 B dynamic LDS > 64 KB default cap: raise the limit (host-side
    // attribute set; idempotent and deterministic, not a stream op).
    (void)hipFuncSetAttribute(reinterpret_cast<const void*>(&relpos_kernel),
                              hipFuncAttributeMaxDynamicSharedMemorySize,
                              SMEM_BYTES);

    relpos_kernel<<<dim3(N), dim3(256), SMEM_BYTES, stream>>>(
        asym, resi, ent, sym, tok, W, out, N);
}